// EachSamplePreintegrationModule_4681514353461
// MI455X (gfx1250) — compile-verified
//
// CDNA5 / gfx1250 — IMU preintegration as a 3-pass associative scan.
// Pass 1: per-(batch,chunk) chunk-local transform E=(M,bv,cp)   [TDM-staged input]
// Pass 2: per-batch exclusive prefix over 64 chunk transforms — expressed as
//         5x5 homogeneous-matrix products, executed on the matrix core with
//         chained V_WMMA_F32_16X16X4_F32 (one chain per wave32).
// Pass 3: per-(batch,chunk) re-simulation from prefix state, emit [in|p|q|v]
//
// CDNA5-specific paths: tensor_load_to_lds (TDM 2-D tiles with LDS padding,
// double-buffered on TENSORcnt via s_wait_tensorcnt), v_wmma_f32_16x16x4_f32
// for the scan combine, wave32 sizing, conflict-free padded LDS layout.

#include <hip/hip_runtime.h>
#include <stdint.h>

#define B_      512
#define T_      4096
#define NC      64            // chunks per sequence
#define CL      64            // steps per chunk (NC*CL == T_)
#define SS      8             // steps per TDM stage
#define NSTG    (CL / SS)     // 8 stages
#define BPB     2             // batch rows per block
#define THREADS (BPB * NC)    // 128 (4 wave32 waves)

#define ROW_DW   48                   // SS * 6 dwords per (chunk,stage) row
#define ROW_PAD  51                   // 48 + 48/16 (TDM pad: 1 dword per 16)
#define REG_DW   (NC * ROW_PAD)       // 3264 dwords per batch row per buffer
#define BUF_DW   (BPB * REG_DW)       // 6528
#define SMEM_DW  (2 * BUF_DW)         // double buffer: 13056 dwords
#define SMEM_BYTES (SMEM_DW * 4)      // 52224 B per workgroup

#define DTF   0.005f
#define DT2H  1.25e-5f                // dt*dt/2
#define EPSF  1e-8f
#define CDTF  (64.0f * DTF)           // CL * dt

// ------------------------------------------------------ feature detection --
#if defined(__has_builtin)
#  if __has_builtin(__builtin_amdgcn_tensor_load_to_lds) && \
      __has_builtin(__builtin_amdgcn_s_wait_tensorcnt)
#    define HAVE_TDM 1
#  else
#    define HAVE_TDM 0
#  endif
#  if __has_builtin(__builtin_amdgcn_wmma_f32_16x16x4_f32)
#    define HAVE_WMMA 1
#  else
#    define HAVE_WMMA 0
#  endif
#else
#  define HAVE_TDM 0
#  define HAVE_WMMA 0
#endif

// ---------------------------------------------------------------- TDM path --
#if HAVE_TDM
typedef unsigned int u32x4 __attribute__((ext_vector_type(4)));
typedef int          i32x8 __attribute__((ext_vector_type(8)));
typedef int          i32x4 __attribute__((ext_vector_type(4)));

// 2-D tile: tile_dim0 = 48 x 4B units (192 B = 8 records), tile_dim1 = 64 rows
// (chunks), row stride 384 units (1536 B = CL records). data_size = 4 B.
// pad_enable, interval 16 dwords (code 3), amount 1 dword (code 0):
// LDS row becomes 51 dwords -> conflict-free lane stride.
__device__ __forceinline__ void tdm_load_2d(unsigned lds_byte,
                                            unsigned long long gaddr) {
  u32x4 g0;
  g0.x = 1u;                                            // count=1, user desc
  g0.y = lds_byte;                                      // LDS byte address
  g0.z = (unsigned)(gaddr & 0xFFFFFFFFull);             // global_addr[31:0]
  g0.w = (unsigned)((gaddr >> 32) & 0x01FFFFFFull)      // global_addr[56:32]
       | (2u << 30);                                    // type = 2 ("image")
  i32x8 g1;
  g1[0] = (int)((2u << 16)        // data_size = 4 B
              | (1u << 20)        // pad_enable
              | (3u << 22));      // pad_interval = 16 dwords (pad_amount=0 -> 1)
  g1[1] = (int)(48u << 16);       // tensor_dim0[15:0] = 48 (in [31:16])
  g1[2] = (int)(64u << 16);       // tensor_dim0 hi = 0 | tensor_dim1 lo = 64
  g1[3] = (int)(48u << 16);       // tensor_dim1 hi = 0 | tile_dim0 = 48
  g1[4] = 64;                     // tile_dim1 = 64, tile_dim2 = 0
  g1[5] = 384;                    // tensor_dim0_stride[31:0] = 384 units
  g1[6] = 0;                      // stride0 hi | stride1 lo
  g1[7] = 0;                      // stride1 hi
  i32x4 z4 = {0, 0, 0, 0};        // groups 2/3 unused (2-D tile)
  i32x8 z8 = {0, 0, 0, 0, 0, 0, 0, 0};
  __builtin_amdgcn_tensor_load_to_lds(g0, g1, z4, z4, z8, 0);
}
#endif

// Stage fill: bring (BPB batch rows) x (64 chunks) x (SS steps) records into
// LDS buffer `buf`. TDM: wave 0 issues one descriptor per batch row.
__device__ __forceinline__ void stage_fill(const float* __restrict__ in,
                                           float* __restrict__ smem,
                                           int b0, int s, int buf) {
#if HAVE_TDM
  (void)smem;
  if (threadIdx.x < 32) {   // one wave owns the TDM queue (TENSORcnt per-wave)
#pragma unroll
    for (int bl = 0; bl < BPB; ++bl) {
      unsigned lds = (unsigned)((buf * BUF_DW + bl * REG_DW) * 4);
      unsigned long long g = (unsigned long long)(uintptr_t)in +
          ((unsigned long long)(b0 + bl) * T_ + (unsigned long long)s * SS)
          * 24ull;   // 6 floats / record
      tdm_load_2d(lds, g);
    }
  }
#else
  // Cooperative fallback: coalesced global reads, padded LDS writes.
  const int total = BPB * NC * ROW_DW;   // 6144 dwords
  for (int i = threadIdx.x; i < total; i += THREADS) {
    int bl = i / (NC * ROW_DW);
    int r  = i - bl * (NC * ROW_DW);
    int chunk = r / ROW_DW;
    int k = r - chunk * ROW_DW;
    size_t g = ((size_t)(b0 + bl) * T_ + (size_t)chunk * CL +
                (size_t)s * SS) * 6 + (size_t)k;
    smem[buf * BUF_DW + bl * REG_DW + chunk * ROW_PAD + k + (k >> 4)] = in[g];
  }
#endif
}

__device__ __forceinline__ void stage_wait(bool more) {
#if HAVE_TDM
  if (threadIdx.x < 32) {
    if (more) __builtin_amdgcn_s_wait_tensorcnt(BPB);  // next stage in flight
    else      __builtin_amdgcn_s_wait_tensorcnt(0);
  }
#else
  (void)more;
#endif
}

// ----------------------------------------------------------------- math -----
__device__ __forceinline__ void matmul3(const float M[9], const float A[9],
                                        float O[9]) {
  O[0] = M[0]*A[0] + M[1]*A[3] + M[2]*A[6];
  O[1] = M[0]*A[1] + M[1]*A[4] + M[2]*A[7];
  O[2] = M[0]*A[2] + M[1]*A[5] + M[2]*A[8];
  O[3] = M[3]*A[0] + M[4]*A[3] + M[5]*A[6];
  O[4] = M[3]*A[1] + M[4]*A[4] + M[5]*A[7];
  O[5] = M[3]*A[2] + M[4]*A[5] + M[5]*A[8];
  O[6] = M[6]*A[0] + M[7]*A[3] + M[8]*A[6];
  O[7] = M[6]*A[1] + M[7]*A[4] + M[8]*A[7];
  O[8] = M[6]*A[2] + M[7]*A[5] + M[8]*A[8];
}

__device__ __forceinline__ void matvec3(const float M[9], const float v[3],
                                        float r[3]) {
  r[0] = M[0]*v[0] + M[1]*v[1] + M[2]*v[2];
  r[1] = M[3]*v[0] + M[4]*v[1] + M[5]*v[2];
  r[2] = M[6]*v[0] + M[7]*v[1] + M[8]*v[2];
}

// A = I + sin(th) K + (1-cos(th)) K^2,  Aa = A*a  (Rodrigues, th = |w| dt)
__device__ __forceinline__ void rodrigues(const float w[3], const float a[3],
                                          float A[9], float Aa[3]) {
  float n  = sqrtf(w[0]*w[0] + w[1]*w[1] + w[2]*w[2]);
  float th = n * DTF;
  float iv = 1.0f / (n + EPSF);
  float x = w[0]*iv, y = w[1]*iv, z = w[2]*iv;
  float s, c;
  __sincosf(th, &s, &c);
  float oc = 1.0f - c;
  A[0] = 1.0f - oc*(y*y + z*z); A[1] = oc*x*y - s*z;         A[2] = oc*x*z + s*y;
  A[3] = oc*x*y + s*z;          A[4] = 1.0f - oc*(x*x + z*z); A[5] = oc*y*z - s*x;
  A[6] = oc*x*z - s*y;          A[7] = oc*y*z + s*x;          A[8] = 1.0f - oc*(x*x + y*y);
  matvec3(A, a, Aa);
}

// --------------------------------------------------------------- kernels ----
// Pass 1: chunk-local transform E = (M, bv, cp) per (batch, chunk) lane.
__global__ void __launch_bounds__(THREADS)
k_chunk(const float* __restrict__ in, float* __restrict__ ws1) {
  extern __shared__ float smem[];
  const int b0    = blockIdx.x * BPB;
  const int bl    = threadIdx.x >> 6;
  const int chunk = threadIdx.x & (NC - 1);

  float M[9]  = {1.f,0.f,0.f, 0.f,1.f,0.f, 0.f,0.f,1.f};
  float bv[3] = {0.f,0.f,0.f};
  float cp[3] = {0.f,0.f,0.f};

  stage_fill(in, smem, b0, 0, 0);
  for (int s = 0; s < NSTG; ++s) {
    if (s + 1 < NSTG) stage_fill(in, smem, b0, s + 1, (s + 1) & 1);
    stage_wait(s + 1 < NSTG);
    __syncthreads();
    const float* reg = smem + (s & 1) * BUF_DW + bl * REG_DW + chunk * ROW_PAD;
#pragma unroll
    for (int i = 0; i < SS; ++i) {
      float rec[6];
#pragma unroll
      for (int d = 0; d < 6; ++d) {
        int k = i * 6 + d;
        rec[d] = reg[k + (k >> 4)];        // undo TDM padding
      }
      float A[9], Aa[3], u[3];
      rodrigues(rec, rec + 3, A, Aa);
      matvec3(M, Aa, u);
      cp[0] += bv[0]*DTF + u[0]*DT2H;  cp[1] += bv[1]*DTF + u[1]*DT2H;
      cp[2] += bv[2]*DTF + u[2]*DT2H;
      bv[0] += u[0]*DTF;  bv[1] += u[1]*DTF;  bv[2] += u[2]*DTF;
      float N[9];
      matmul3(M, A, N);
#pragma unroll
      for (int q = 0; q < 9; ++q) M[q] = N[q];
    }
    __syncthreads();
  }
  float* o = ws1 + ((size_t)(b0 + bl) * NC + chunk) * 16;
#pragma unroll
  for (int q = 0; q < 9; ++q) o[q] = M[q];
  o[9]  = bv[0]; o[10] = bv[1]; o[11] = bv[2];
  o[12] = cp[0]; o[13] = cp[1]; o[14] = cp[2];
}

// Pass 2: per-batch exclusive prefix over the NC chunk transforms.
//
// Scan element embedded as a 5x5 homogeneous matrix
//   S = [ M  bv  cp ; 0 1 n*dt ; 0 0 1 ],  composition == matrix product.
// One chain per wave32. We iterate the *transposed* prefix Q = S_acc^T via
//   Q_{j+1} = E_j^T * Q_j
// so the iterated operand lives in the B/D position: D -> next B is a
// half-wave shuffle (D rows 0..4 sit in d0..d4, lanes 0..15 = columns).
// Each product: K = 8 (>= 5) via two chained V_WMMA_F32_16X16X4_F32.
#if HAVE_WMMA
typedef float v2f __attribute__((ext_vector_type(2)));
typedef float v8f __attribute__((ext_vector_type(8)));

__global__ void __launch_bounds__(32)
k_prefix(const float* __restrict__ ws1, float* __restrict__ ws2) {
  const int  b    = blockIdx.x;          // one chain per wave
  const int  lane = threadIdx.x;
  const int  r    = lane & 15;
  const bool hi   = lane >= 16;

  // Q0 = I (5x5) in B-operand layout:
  //   b0: K=0 (lo lanes) / K=2 (hi), b1: K=1 / K=3, b2: K=4 / K=6(=0)
  float b0 = (!hi && r == 0) ? 1.f : ((hi && r == 2) ? 1.f : 0.f);
  float b1 = (!hi && r == 1) ? 1.f : ((hi && r == 3) ? 1.f : 0.f);
  float b2 = (!hi && r == 4) ? 1.f : 0.f;

  for (int j = 0; j < NC; ++j) {
    const float* e = ws1 + ((size_t)b * NC + j) * 16;
    float*       o = ws2 + ((size_t)b * NC + j) * 16;

    // Emit exclusive prefix: S_acc[row][col] = Q[col][row].
    //  lane c (lo, c<3): b0=M[c][0], b1=M[c][1], b2=cp[c]
    //  lane 16+c (c<3) : b0=M[c][2], b1=bv[c]
    if (!hi && r < 3) {
      o[3*r + 0] = b0;
      o[3*r + 1] = b1;
      o[12 + r]  = b2;
    } else if (hi && r < 3) {
      o[3*r + 2] = b0;
      o[9 + r]   = b1;
    }

    // Build A = E_j^T as two 16x4 operands: A(row=r, k) = S_j[k][r].
    //  S_j rows 0..2 = [M | be | ce], row 3 = [0,0,0,1,CDT], row 4 = e5.
    float a0 = 0.f, a1 = 0.f, a2 = 0.f;
    if (r <= 4) {
      int i0 = hi ? (r < 3 ? 6 + r : (r == 3 ? 11 : 14))    // k=2
                  : (r < 3 ? r     : (r == 3 ? 9  : 12));   // k=0
      a0 = e[i0];
      if (!hi) a1 = e[r < 3 ? 3 + r : (r == 3 ? 10 : 13)];  // k=1
    }
    if (hi)            a1 = (r == 3) ? 1.f : ((r == 4) ? CDTF : 0.f); // k=3
    if (!hi && r == 4) a2 = 1.f;                                      // k=4

    v2f A1 = {a0, a1};
    v2f A2 = {a2, 0.f};
    v2f B1 = {b0, b1};
    v2f B2 = {b2, 0.f};
    v8f acc = {0.f, 0.f, 0.f, 0.f, 0.f, 0.f, 0.f, 0.f};
    acc = __builtin_amdgcn_wmma_f32_16x16x4_f32(false, A1, false, B1,
                                                (short)0, acc, false, false);
    acc = __builtin_amdgcn_wmma_f32_16x16x4_f32(false, A2, false, B2,
                                                (short)0, acc, false, false);

    // D -> B layout for next iteration (rows 2,3 move to hi lane halves).
    float d2s = __shfl(acc[2], r, 32);
    float d3s = __shfl(acc[3], r, 32);
    b0 = hi ? d2s : acc[0];
    b1 = hi ? d3s : acc[1];
    b2 = hi ? 0.f : acc[4];
  }
}
#else
__global__ void k_prefix(const float* __restrict__ ws1,
                         float* __restrict__ ws2) {
  int b = blockIdx.x * blockDim.x + threadIdx.x;
  if (b >= B_) return;
  float M[9]  = {1.f,0.f,0.f, 0.f,1.f,0.f, 0.f,0.f,1.f};
  float bv[3] = {0.f,0.f,0.f};
  float cp[3] = {0.f,0.f,0.f};
  for (int j = 0; j < NC; ++j) {
    const float* e = ws1 + ((size_t)b * NC + j) * 16;
    float*       o = ws2 + ((size_t)b * NC + j) * 16;
#pragma unroll
    for (int q = 0; q < 9; ++q) o[q] = M[q];
    o[9]  = bv[0]; o[10] = bv[1]; o[11] = bv[2];
    o[12] = cp[0]; o[13] = cp[1]; o[14] = cp[2];
    float Me[9], be[3], ce[3], mb[3], mc[3];
#pragma unroll
    for (int q = 0; q < 9; ++q) Me[q] = e[q];
    be[0] = e[9];  be[1] = e[10]; be[2] = e[11];
    ce[0] = e[12]; ce[1] = e[13]; ce[2] = e[14];
    matvec3(M, ce, mc);
    matvec3(M, be, mb);
    cp[0] += bv[0]*CDTF + mc[0];  cp[1] += bv[1]*CDTF + mc[1];
    cp[2] += bv[2]*CDTF + mc[2];
    bv[0] += mb[0];  bv[1] += mb[1];  bv[2] += mb[2];
    float N[9];
    matmul3(M, Me, N);
#pragma unroll
    for (int q = 0; q < 9; ++q) M[q] = N[q];
  }
}
#endif

// Pass 3: re-simulate each chunk from its prefix state and emit 16 channels.
__global__ void __launch_bounds__(THREADS)
k_emit(const float* __restrict__ in, const float* __restrict__ ws2,
       float* __restrict__ out) {
  extern __shared__ float smem[];
  const int b0    = blockIdx.x * BPB;
  const int bl    = threadIdx.x >> 6;
  const int chunk = threadIdx.x & (NC - 1);

  const float* st = ws2 + ((size_t)(b0 + bl) * NC + chunk) * 16;
  float R[9], v[3], p[3];
#pragma unroll
  for (int q = 0; q < 9; ++q) R[q] = st[q];
  v[0] = st[9];  v[1] = st[10]; v[2] = st[11];
  p[0] = st[12]; p[1] = st[13]; p[2] = st[14];

  const size_t tbase = (size_t)(b0 + bl) * T_ + (size_t)chunk * CL;

  stage_fill(in, smem, b0, 0, 0);
  for (int s = 0; s < NSTG; ++s) {
    if (s + 1 < NSTG) stage_fill(in, smem, b0, s + 1, (s + 1) & 1);
    stage_wait(s + 1 < NSTG);
    __syncthreads();
    const float* reg = smem + (s & 1) * BUF_DW + bl * REG_DW + chunk * ROW_PAD;
#pragma unroll
    for (int i = 0; i < SS; ++i) {
      float rec[6];
#pragma unroll
      for (int d = 0; d < 6; ++d) {
        int k = i * 6 + d;
        rec[d] = reg[k + (k >> 4)];
      }
      float A[9], Aa[3], u[3];
      rodrigues(rec, rec + 3, A, Aa);
      matvec3(R, Aa, u);                       // u = dR_new * a
      p[0] += v[0]*DTF + u[0]*DT2H;  p[1] += v[1]*DTF + u[1]*DT2H;
      p[2] += v[2]*DTF + u[2]*DT2H;
      v[0] += u[0]*DTF;  v[1] += u[1]*DTF;  v[2] += u[2]*DTF;
      float N[9];
      matmul3(R, A, N);
#pragma unroll
      for (int q = 0; q < 9; ++q) R[q] = N[q];

      // R -> quaternion without arccos: algebraic in x = clip((tr-1)/2):
      // sin(angle)=sqrt(1-x^2), cos(h)=sqrt((1+x)/2), sin(h)=sqrt((1-x)/2).
      float tr = R[0] + R[4] + R[8];
      float xc = fminf(fmaxf(0.5f * (tr - 1.0f), -1.0f + 1e-7f), 1.0f - 1e-7f);
      float sa = sqrtf(fmaxf(1.0f - xc * xc, 0.0f));
      float i2 = 1.0f / (2.0f * sa + EPSF);
      float ax = (R[7] - R[5]) * i2;
      float ay = (R[2] - R[6]) * i2;
      float az = (R[3] - R[1]) * i2;
      float ch = sqrtf(0.5f * (1.0f + xc));
      float sh = sqrtf(0.5f * (1.0f - xc));

      size_t t = tbase + (size_t)s * SS + (size_t)i;
      float4* ob = (float4*)(out + t * 16);    // 64 B record, 16 B aligned
      ob[0] = make_float4(rec[0], rec[1], rec[2], rec[3]);
      ob[1] = make_float4(rec[4], rec[5], p[0], p[1]);
      ob[2] = make_float4(p[2], ch, ax * sh, ay * sh);
      ob[3] = make_float4(az * sh, v[0], v[1], v[2]);
    }
    __syncthreads();
  }
}

// ----------------------------------------------------------------- launch ---
extern "C" void kernel_launch(void* const* d_in, const int* in_sizes, int n_in,
                              void* d_out, int out_size, void* d_ws,
                              size_t ws_size, hipStream_t stream) {
  (void)in_sizes; (void)n_in; (void)out_size; (void)ws_size;
  const float* in  = (const float*)d_in[0];      // (512, 4096, 6) f32
  float*       out = (float*)d_out;              // (512, 4096, 16) f32
  float* ws1 = (float*)d_ws;                     // 512*64*16 f32 = 2 MB
  float* ws2 = ws1 + (size_t)B_ * NC * 16;       // 2 MB more

  k_chunk <<<B_ / BPB, THREADS, SMEM_BYTES, stream>>>(in, ws1);
  k_prefix<<<B_, 32, 0, stream>>>(ws1, ws2);     // one wave per chain
  k_emit  <<<B_ / BPB, THREADS, SMEM_BYTES, stream>>>(in, ws2, out);
}